// local_pool_pn_37443524887128
// MI455X (gfx1250) — compile-verified
//
#include <hip/hip_runtime.h>
#include <cstdint>
#include <cstddef>

// ---------------------------------------------------------------------------
// local_pool_pointnet for MI455X (gfx1250, wave32)
//  - per-point MLP GEMMs run on V_WMMA_F32_16X16X4_F32 (exact f32 math)
//  - scatter-mean via global f32 atomics into d_out reused as grid scratch
//  - bandwidth-bound problem (~2.3 GB traffic @ 23.3 TB/s); compute fused
// ---------------------------------------------------------------------------

#define RESO   64
#define G3     (RESO * RESO * RESO)      // 262144 cells
#define MID    32
#define CAT    64                        // 2*mid
#define NPTS   65536
#define BS     8
#define NB     5
#define PTOT   (BS * NPTS)               // 524288 points

#define TILE_PTS      64                 // points per workgroup
#define WAVES_PER_BLK 4                  // 16 points per wave (one WMMA M-tile)
#define BLK_THREADS   (WAVES_PER_BLK * 32)
#define CAT_STRIDE    66                 // padded LDS row stride (even -> float2 ok)

typedef float v2f __attribute__((ext_vector_type(2)));
typedef float v8f __attribute__((ext_vector_type(8)));

__device__ __forceinline__ v8f wmma_f32(v2f a, v2f b, v8f c) {
  // D = A(16x4) * B(4x16) + C, exact f32, RNE
  return __builtin_amdgcn_wmma_f32_16x16x4_f32(false, a, false, b, (short)0, c,
                                               false, false);
}

__device__ __forceinline__ int point_cell(float px, float py, float pz) {
  // q = clip(p/1.101 + 0.5, 0, 1-1e-3); g = floor(q*64); idx = x + 64*(y+64*z)
  float qx = fminf(fmaxf(px / 1.101f + 0.5f, 0.0f), 0.999f);
  float qy = fminf(fmaxf(py / 1.101f + 0.5f, 0.0f), 0.999f);
  float qz = fminf(fmaxf(pz / 1.101f + 0.5f, 0.0f), 0.999f);
  int gx = (int)(qx * (float)RESO);
  int gy = (int)(qy * (float)RESO);
  int gz = (int)(qz * (float)RESO);
  return gx + RESO * (gy + RESO * gz);
}

// ---------------------------------------------------------------------------
// Fused residual block: builds the 64-wide input (block0: p@l1_w+b fused;
// later blocks: [x, grid-mean gather]), then 3 WMMA GEMMs:
//   t1 = relu(relu(cat) @ w1 + b1);  out = (t1 @ w2 + b2) + cat @ w3
// ---------------------------------------------------------------------------
template <bool FIRST>
__global__ __launch_bounds__(BLK_THREADS) void resblock_kernel(
    const float* p, const float* l1w, const float* l1b,
    const float* x_in, const float* gridsum, const float* cnt,
    int* idx,                                   // written if FIRST, read else
    const float* w1, const float* b1, const float* w2, const float* b2,
    const float* w3, float* x_out) {
  __shared__ float sCat[TILE_PTS * CAT_STRIDE];        // 16.9 KB
  __shared__ float sT1[WAVES_PER_BLK][16 * MID];       // 8 KB
  __shared__ float sW1[CAT * MID];                     // 8 KB
  __shared__ float sW2[MID * MID];                     // 4 KB
  __shared__ float sW3[CAT * MID];                     // 8 KB
  __shared__ float sB1[MID];
  __shared__ float sB2[MID];
  __shared__ float sAux[TILE_PTS * 4];                 // FIRST: xyz rows, else: max(cnt,1)
  __shared__ int   sCell[TILE_PTS];
  __shared__ float sL1[3 * CAT];
  __shared__ float sLB[CAT];

  const int tid = threadIdx.x;
  const int pt0 = blockIdx.x * TILE_PTS;
  const int bb  = pt0 / NPTS;

  for (int i = tid; i < CAT * MID; i += BLK_THREADS) { sW1[i] = w1[i]; sW3[i] = w3[i]; }
  for (int i = tid; i < MID * MID; i += BLK_THREADS) sW2[i] = w2[i];
  if (tid < MID) { sB1[tid] = b1[tid]; sB2[tid] = b2[tid]; }

  if (FIRST) {
    for (int i = tid; i < 3 * CAT; i += BLK_THREADS) sL1[i] = l1w[i];
    if (tid < CAT) sLB[tid] = l1b[tid];
    for (int i = tid; i < TILE_PTS * 3; i += BLK_THREADS)
      sAux[i] = p[(size_t)pt0 * 3 + i];
    __syncthreads();
    if (tid < TILE_PTS) {
      int cell = point_cell(sAux[tid * 3 + 0], sAux[tid * 3 + 1], sAux[tid * 3 + 2]);
      sCell[tid] = cell;
      idx[pt0 + tid] = cell;
    }
    __syncthreads();
    for (int i = tid; i < TILE_PTS * CAT; i += BLK_THREADS) {
      int row = i >> 6, col = i & (CAT - 1);
      float v = sLB[col] + sAux[row * 3 + 0] * sL1[0 * CAT + col]
                         + sAux[row * 3 + 1] * sL1[1 * CAT + col]
                         + sAux[row * 3 + 2] * sL1[2 * CAT + col];
      sCat[row * CAT_STRIDE + col] = v;
    }
  } else {
    if (tid < TILE_PTS) {
      int cell = idx[pt0 + tid];
      sCell[tid] = cell;
      sAux[tid] = fmaxf(cnt[(size_t)bb * G3 + cell], 1.0f);
    }
    __syncthreads();
    for (int i = tid; i < TILE_PTS * MID; i += BLK_THREADS) {
      int row = i >> 5, col = i & (MID - 1);
      sCat[row * CAT_STRIDE + col] = x_in[((size_t)(pt0 + row)) * MID + col];
      sCat[row * CAT_STRIDE + MID + col] =
          gridsum[((size_t)bb * G3 + sCell[row]) * MID + col] / sAux[row];
    }
  }
  __syncthreads();

  // -------- WMMA region: all 128 lanes active (EXEC all-ones) --------
  const int lane = tid & 31;
  const int wave = tid >> 5;
  const int mn   = lane & 15;              // row for A, col for B/C
  const int koff = (lane >> 4) << 1;       // K pair offset within k-step
  const int rb   = (lane >> 4) << 3;       // C/D row base: 0 or 8
  const float* catRow = &sCat[(wave * 16 + mn) * CAT_STRIDE];

  v8f acc1a, acc1b;
  {
    float va = sB1[mn], vb = sB1[16 + mn];
#pragma unroll
    for (int r = 0; r < 8; ++r) { acc1a[r] = va; acc1b[r] = vb; }
  }
#pragma unroll
  for (int kk = 0; kk < CAT; kk += 4) {    // relu(cat) @ w1
    v2f a, ba, bbv;
    a.x  = fmaxf(catRow[kk + koff + 0], 0.0f);
    a.y  = fmaxf(catRow[kk + koff + 1], 0.0f);
    ba.x = sW1[(kk + koff + 0) * MID + mn];
    ba.y = sW1[(kk + koff + 1) * MID + mn];
    bbv.x = sW1[(kk + koff + 0) * MID + 16 + mn];
    bbv.y = sW1[(kk + koff + 1) * MID + 16 + mn];
    acc1a = wmma_f32(a, ba, acc1a);
    acc1b = wmma_f32(a, bbv, acc1b);
  }
  {
    float* t1 = sT1[wave];                 // C-layout -> row-major LDS (relu'd)
#pragma unroll
    for (int r = 0; r < 8; ++r) {
      t1[(rb + r) * MID + mn]      = fmaxf(acc1a[r], 0.0f);
      t1[(rb + r) * MID + 16 + mn] = fmaxf(acc1b[r], 0.0f);
    }
  }
  __syncthreads();

  v8f acc2a, acc2b;
  {
    float va = sB2[mn], vb = sB2[16 + mn];
#pragma unroll
    for (int r = 0; r < 8; ++r) { acc2a[r] = va; acc2b[r] = vb; }
  }
  const float* t1row = &sT1[wave][mn * MID];
#pragma unroll
  for (int kk = 0; kk < MID; kk += 4) {    // t1 @ w2
    v2f a, ba, bbv;
    a.x  = t1row[kk + koff + 0];
    a.y  = t1row[kk + koff + 1];
    ba.x = sW2[(kk + koff + 0) * MID + mn];
    ba.y = sW2[(kk + koff + 1) * MID + mn];
    bbv.x = sW2[(kk + koff + 0) * MID + 16 + mn];
    bbv.y = sW2[(kk + koff + 1) * MID + 16 + mn];
    acc2a = wmma_f32(a, ba, acc2a);
    acc2b = wmma_f32(a, bbv, acc2b);
  }
#pragma unroll
  for (int kk = 0; kk < CAT; kk += 4) {    // + cat @ w3 (shortcut, no relu)
    v2f a, ba, bbv;
    a.x  = catRow[kk + koff + 0];
    a.y  = catRow[kk + koff + 1];
    ba.x = sW3[(kk + koff + 0) * MID + mn];
    ba.y = sW3[(kk + koff + 1) * MID + mn];
    bbv.x = sW3[(kk + koff + 0) * MID + 16 + mn];
    bbv.y = sW3[(kk + koff + 1) * MID + 16 + mn];
    acc2a = wmma_f32(a, ba, acc2a);
    acc2b = wmma_f32(a, bbv, acc2b);
  }
#pragma unroll
  for (int r = 0; r < 8; ++r) {
    size_t pt = (size_t)(pt0 + wave * 16 + rb + r);
    x_out[pt * MID + mn]      = acc2a[r];
    x_out[pt * MID + 16 + mn] = acc2b[r];
  }
}

// ---------------------------------------------------------------------------
__global__ __launch_bounds__(256) void zero_kernel(float4* __restrict__ a, size_t n4) {
  size_t i = (size_t)blockIdx.x * blockDim.x + threadIdx.x;
  const size_t stride = (size_t)gridDim.x * blockDim.x;
  const float4 z = make_float4(0.f, 0.f, 0.f, 0.f);
  for (; i < n4; i += stride) a[i] = z;
}

// scatter-sum of x[P,32] into grid[bs,G,32] + counts (1 thread per (pt,f))
__global__ __launch_bounds__(256) void scatter_kernel(
    const float* __restrict__ x, const int* __restrict__ idx,
    float* __restrict__ gridsum, float* __restrict__ cnt) {
  size_t t = (size_t)blockIdx.x * 256 + threadIdx.x;
  int pt = (int)(t >> 5);
  int f  = (int)(t & 31);
  int b  = pt / NPTS;
  int cell = idx[pt];
  atomicAdd(&gridsum[((size_t)b * G3 + cell) * MID + f], x[t]);
  if (f == 0) atomicAdd(&cnt[(size_t)b * G3 + cell], 1.0f);
}

// fused c = x@l2_w + l2_b scattered straight into transposed out[b,f,cell]
__global__ __launch_bounds__(256) void final_scatter_kernel(
    const float* __restrict__ x, const int* __restrict__ idx,
    const float* __restrict__ l2w, const float* __restrict__ l2b,
    float* __restrict__ out, float* __restrict__ cnt) {
  __shared__ float sW[MID * MID];
  __shared__ float sB[MID];
  __shared__ float sX[8 * MID];
  __shared__ int   sC[8];
  const int tid = threadIdx.x;
  const int pt0 = blockIdx.x * 8;
  for (int i = tid; i < MID * MID; i += 256) sW[i] = l2w[i];
  if (tid < MID) sB[tid] = l2b[tid];
  sX[tid] = x[(size_t)pt0 * MID + tid];
  if (tid < 8) sC[tid] = idx[pt0 + tid];
  __syncthreads();
  const int lp = tid >> 5;
  const int f  = tid & 31;
  float c = sB[f];
#pragma unroll
  for (int k = 0; k < MID; ++k) c = fmaf(sX[lp * MID + k], sW[k * MID + f], c);
  const int pt = pt0 + lp;
  const int b = pt / NPTS;
  const int cell = sC[lp];
  atomicAdd(&out[((size_t)b * MID + f) * G3 + cell], c);
  if (f == 0) atomicAdd(&cnt[(size_t)b * G3 + cell], 1.0f);
}

__global__ __launch_bounds__(256) void normalize_kernel(
    float* __restrict__ out, const float* __restrict__ cnt) {
  size_t t = (size_t)blockIdx.x * 256 + threadIdx.x;
  size_t cell = t & (size_t)(G3 - 1);
  size_t b = t >> 23;  // / (MID*G3) = 2^23
  out[t] = out[t] / fmaxf(cnt[b * G3 + cell], 1.0f);
}

// ---------------------------------------------------------------------------
extern "C" void kernel_launch(void* const* d_in, const int* in_sizes, int n_in,
                              void* d_out, int out_size, void* d_ws, size_t ws_size,
                              hipStream_t stream) {
  (void)in_sizes; (void)n_in; (void)out_size; (void)ws_size;
  const float* p    = (const float*)d_in[0];
  const float* l1w  = (const float*)d_in[1];
  const float* l1b  = (const float*)d_in[2];
  const float* fc1w = (const float*)d_in[3];
  const float* fc1b = (const float*)d_in[4];
  const float* fc2w = (const float*)d_in[5];
  const float* fc2b = (const float*)d_in[6];
  const float* fc3w = (const float*)d_in[7];
  const float* l2w  = (const float*)d_in[8];
  const float* l2b  = (const float*)d_in[9];
  float* out = (float*)d_out;

  // workspace: idx (2MB) | cnt (8MB) | x ping (64MB) | x pong (64MB) = 138MB
  char* ws = (char*)d_ws;
  int*   idx  = (int*)ws;
  float* cnt  = (float*)(ws + (size_t)PTOT * 4);
  float* xb0  = (float*)(ws + (size_t)PTOT * 4 + (size_t)BS * G3 * 4);
  float* xb1  = xb0 + (size_t)PTOT * MID;
  float* xbuf[2] = {xb0, xb1};

  const dim3 rb_grid(PTOT / TILE_PTS), rb_blk(BLK_THREADS);
  const size_t grid_f = (size_t)BS * MID * G3;   // == out_size
  const size_t cnt_f  = (size_t)BS * G3;

  resblock_kernel<true><<<rb_grid, rb_blk, 0, stream>>>(
      p, l1w, l1b, nullptr, nullptr, nullptr, idx,
      fc1w, fc1b, fc2w, fc2b, fc3w, xbuf[0]);

  int cur = 0;
  for (int i = 1; i < NB; ++i) {
    zero_kernel<<<2048, 256, 0, stream>>>((float4*)out, grid_f / 4);
    zero_kernel<<<512, 256, 0, stream>>>((float4*)cnt, cnt_f / 4);
    scatter_kernel<<<(PTOT * MID) / 256, 256, 0, stream>>>(xbuf[cur], idx, out, cnt);
    resblock_kernel<false><<<rb_grid, rb_blk, 0, stream>>>(
        nullptr, nullptr, nullptr, xbuf[cur], out, cnt, idx,
        fc1w + (size_t)i * CAT * MID, fc1b + (size_t)i * MID,
        fc2w + (size_t)i * MID * MID, fc2b + (size_t)i * MID,
        fc3w + (size_t)i * CAT * MID, xbuf[1 - cur]);
    cur = 1 - cur;
  }

  zero_kernel<<<2048, 256, 0, stream>>>((float4*)out, grid_f / 4);
  zero_kernel<<<512, 256, 0, stream>>>((float4*)cnt, cnt_f / 4);
  final_scatter_kernel<<<PTOT / 8, 256, 0, stream>>>(xbuf[cur], idx, l2w, l2b, out, cnt);
  normalize_kernel<<<(unsigned)(grid_f / 256), 256, 0, stream>>>(out, cnt);
}